// PinSagge_4509715661236
// MI455X (gfx1250) — compile-verified
//
#include <hip/hip_runtime.h>
#include <hip/hip_bf16.h>
#include <math.h>

typedef float v2f __attribute__((ext_vector_type(2)));
typedef float v8f __attribute__((ext_vector_type(8)));

#define N_NODES 12000
#define KNBR    32
#define CDIM    64
#define HDIM    64
#define ALPHA   0.3f

// ---------------------------------------------------------------------------
// Kernel 1: gather per-(node,neighbor) edge weights + per-node weight sums.
// One wave (32 lanes) per node; reused by both layers.
// ---------------------------------------------------------------------------
__global__ void k_gather_w(const float* __restrict__ W,
                           const int* __restrict__ nbr,
                           float* __restrict__ nb_w,
                           float* __restrict__ wsum) {
    int lane = threadIdx.x & 31;
    int wave = threadIdx.x >> 5;
    int n = blockIdx.x * 8 + wave;
    if (n >= N_NODES) return;
    int g = nbr[n * KNBR + lane];
    float w = W[(size_t)n * N_NODES + (size_t)g];
    nb_w[n * KNBR + lane] = w;
    float s = w;
#pragma unroll
    for (int d = 16; d >= 1; d >>= 1) s += __shfl_xor(s, d, 32);
    if (lane == 0) wsum[n] = s;
}

// ---------------------------------------------------------------------------
// Kernel 2: agg[n,:] = sum_k w_k * leaky(emb[nbr[n,k],:] @ Qw + Qb) / (sum w + eps)
// One wave per node; two 16-row A tiles processed sequentially (t-loop).
// The k-step loop is kept ROLLED (unroll 2) so the sB LDS fragment loads
// depend on the induction variable and cannot be LICM-hoisted out of the
// t-loop (which previously materialized all 64 B fragments -> VGPR spill).
// Live set per iteration: acc[4] (32 VGPRs) + ~8 fragment VGPRs. No scratch.
// ---------------------------------------------------------------------------
__global__ void k_aggregate(const float* __restrict__ emb,
                            const float* __restrict__ Qw,
                            const float* __restrict__ Qb,
                            const int* __restrict__ nbr,
                            const float* __restrict__ nb_w,
                            const float* __restrict__ wsum,
                            float* __restrict__ agg) {
    __shared__ float sB[4 * 16 * 32 * 2];  // 16 KB, B fragments for Qw

    // Stage Qw swizzled: fragment (c,s), lane ln holds {Qw[k0][col], Qw[k0+1][col]}
    // with k0 = 4s + (ln<16 ? 0 : 2), col = c*16 + (ln&15).
    for (int i = threadIdx.x; i < 4 * 16 * 32; i += 256) {
        int c   = i >> 9;
        int rem = i & 511;
        int s   = rem >> 5;
        int ln  = rem & 31;
        int k0  = 4 * s + ((ln < 16) ? 0 : 2);
        int col = c * 16 + (ln & 15);
        sB[i * 2 + 0] = Qw[k0 * HDIM + col];
        sB[i * 2 + 1] = Qw[(k0 + 1) * HDIM + col];
    }
    __syncthreads();

    int lane = threadIdx.x & 31;
    int wave = threadIdx.x >> 5;
    int l16  = lane & 15;
    int hoff = (lane < 16) ? 0 : 2;  // K sub-offset for A/B fragments
    int n = blockIdx.x * 8 + wave;
    if (n >= N_NODES) return;        // whole wave exits together (EXEC stays full)

    int   g  = nbr[n * KNBR + lane];
    float wv = nb_w[n * KNBR + lane];

    float part[4] = {0.0f, 0.0f, 0.0f, 0.0f};
    float qb[4];
#pragma unroll
    for (int c = 0; c < 4; ++c) qb[c] = Qb[c * 16 + l16];

#pragma unroll 1
    for (int t = 0; t < 2; ++t) {
        // A-tile rows: lane holds row l16 of tile t (neighbor t*16 + l16)
        int rt = __shfl(g, t * 16 + l16, 32);
        const float* pt = emb + (size_t)rt * CDIM + hoff;

        v8f acc[4];
#pragma unroll
        for (int c = 0; c < 4; ++c)
#pragma unroll
            for (int j = 0; j < 8; ++j) acc[c][j] = 0.0f;

#pragma unroll 2
        for (int s = 0; s < 16; ++s) {
            v2f a = *(const v2f*)(pt + 4 * s);
#pragma unroll
            for (int c = 0; c < 4; ++c) {
                v2f b = *(const v2f*)&sB[((c * 16 + s) * 32 + lane) * 2];
                acc[c] = __builtin_amdgcn_wmma_f32_16x16x4_f32(
                    false, a, false, b, (short)0, acc[c], false, false);
            }
        }

        // bias + leaky + weighted row reduction for this tile
#pragma unroll
        for (int c = 0; c < 4; ++c) {
#pragma unroll
            for (int j = 0; j < 8; ++j) {
                float v = acc[c][j] + qb[c];
                v = (v >= 0.0f) ? v : ALPHA * v;
                int row = t * 16 + j + ((lane < 16) ? 0 : 8);  // C layout: M=j / j+8
                part[c] += v * __shfl(wv, row, 32);
            }
        }
    }

    float inv = 1.0f / (wsum[n] + 1e-6f);
#pragma unroll
    for (int c = 0; c < 4; ++c) {
        float p = part[c] + __shfl_xor(part[c], 16, 32);  // combine row halves
        if (lane < 16) agg[(size_t)n * HDIM + c * 16 + l16] = p * inv;
    }
}

// ---------------------------------------------------------------------------
// Kernel 3: out[n,:] = normalize(leaky([x[n,:], agg[n,:]] @ Ww + Wb))
// One wave per 16 nodes. K=128 -> 32 k-steps (16 from x, 16 from agg).
// No outer loop -> no LICM hazard; fully unrolled, acc[4] only.
// ---------------------------------------------------------------------------
__global__ void k_dense(const float* __restrict__ x,
                        const float* __restrict__ agg,
                        const float* __restrict__ Ww,
                        const float* __restrict__ Wb,
                        float* __restrict__ out) {
    __shared__ float sB[4 * 32 * 32 * 2];  // 32 KB, B fragments for Ww (128x64)

    for (int i = threadIdx.x; i < 4 * 32 * 32; i += 256) {
        int c   = i >> 10;
        int rem = i & 1023;
        int s   = rem >> 5;
        int ln  = rem & 31;
        int k0  = 4 * s + ((ln < 16) ? 0 : 2);
        int col = c * 16 + (ln & 15);
        sB[i * 2 + 0] = Ww[k0 * HDIM + col];
        sB[i * 2 + 1] = Ww[(k0 + 1) * HDIM + col];
    }
    __syncthreads();

    int lane = threadIdx.x & 31;
    int wave = threadIdx.x >> 5;
    int l16  = lane & 15;
    int hoff = (lane < 16) ? 0 : 2;
    int nb = (blockIdx.x * 8 + wave) * 16;  // first node of this wave's tile
    if (nb >= N_NODES) return;

    int rowA = nb + l16;
    if (rowA >= N_NODES) rowA = N_NODES - 1;  // clamp (stores are guarded below)
    const float* px = x   + (size_t)rowA * CDIM + hoff;
    const float* pa = agg + (size_t)rowA * HDIM + hoff;

    v8f acc[4];
#pragma unroll
    for (int c = 0; c < 4; ++c)
#pragma unroll
        for (int j = 0; j < 8; ++j) acc[c][j] = 0.0f;

    for (int s = 0; s < 16; ++s) {  // K = 0..63 from x
        v2f a = *(const v2f*)(px + 4 * s);
#pragma unroll
        for (int c = 0; c < 4; ++c) {
            v2f b = *(const v2f*)&sB[((c * 32 + s) * 32 + lane) * 2];
            acc[c] = __builtin_amdgcn_wmma_f32_16x16x4_f32(
                false, a, false, b, (short)0, acc[c], false, false);
        }
    }
    for (int s = 0; s < 16; ++s) {  // K = 64..127 from agg
        v2f a = *(const v2f*)(pa + 4 * s);
#pragma unroll
        for (int c = 0; c < 4; ++c) {
            v2f b = *(const v2f*)&sB[((c * 32 + 16 + s) * 32 + lane) * 2];
            acc[c] = __builtin_amdgcn_wmma_f32_16x16x4_f32(
                false, a, false, b, (short)0, acc[c], false, false);
        }
    }

    // bias + leaky (in C-fragment layout)
#pragma unroll
    for (int c = 0; c < 4; ++c) {
        float wb = Wb[c * 16 + l16];
#pragma unroll
        for (int j = 0; j < 8; ++j) {
            float v = acc[c][j] + wb;
            acc[c][j] = (v >= 0.0f) ? v : ALPHA * v;
        }
    }

    // per-row L2 norm: row (j, lane-half) spread over the 16 lanes of each half
    float invn[8];
#pragma unroll
    for (int j = 0; j < 8; ++j) {
        float ss = 0.0f;
#pragma unroll
        for (int c = 0; c < 4; ++c) ss += acc[c][j] * acc[c][j];
        ss += __shfl_xor(ss, 1, 32);
        ss += __shfl_xor(ss, 2, 32);
        ss += __shfl_xor(ss, 4, 32);
        ss += __shfl_xor(ss, 8, 32);
        invn[j] = 1.0f / (sqrtf(ss) + 1e-6f);
    }

#pragma unroll
    for (int c = 0; c < 4; ++c)
#pragma unroll
        for (int j = 0; j < 8; ++j) {
            int node = nb + j + ((lane < 16) ? 0 : 8);
            if (node < N_NODES)
                out[(size_t)node * HDIM + c * 16 + l16] = acc[c][j] * invn[j];
        }
}

// ---------------------------------------------------------------------------
// Launch: 2 layers. nb_w / wsum computed once and reused.
// ---------------------------------------------------------------------------
extern "C" void kernel_launch(void* const* d_in, const int* in_sizes, int n_in,
                              void* d_out, int out_size, void* d_ws, size_t ws_size,
                              hipStream_t stream) {
    const float* emb = (const float*)d_in[0];
    const float* W   = (const float*)d_in[1];
    const int*   nbr = (const int*)d_in[2];
    const float* Q0w = (const float*)d_in[3];
    const float* Q0b = (const float*)d_in[4];
    const float* W0w = (const float*)d_in[5];
    const float* W0b = (const float*)d_in[6];
    const float* Q1w = (const float*)d_in[7];
    const float* Q1b = (const float*)d_in[8];
    const float* W1w = (const float*)d_in[9];
    const float* W1b = (const float*)d_in[10];
    float* out = (float*)d_out;

    float* ws   = (float*)d_ws;
    float* nb_w = ws;                          // N*K        = 384000
    float* wsum = ws + 384000;                 // N          =  12000
    float* aggb = ws + 396000;                 // N*H        = 768000
    float* x1   = ws + 396000 + 768000;        // N*H        = 768000

    dim3 blk(256);
    const int gridN  = (N_NODES + 7) / 8;                 // 1 node / wave
    const int gridD  = ((N_NODES + 15) / 16 + 7) / 8;     // 16 nodes / wave

    k_gather_w <<<gridN, blk, 0, stream>>>(W, nbr, nb_w, wsum);
    // Layer 0
    k_aggregate<<<gridN, blk, 0, stream>>>(emb, Q0w, Q0b, nbr, nb_w, wsum, aggb);
    k_dense    <<<gridD, blk, 0, stream>>>(emb, aggb, W0w, W0b, x1);
    // Layer 1
    k_aggregate<<<gridN, blk, 0, stream>>>(x1, Q1w, Q1b, nbr, nb_w, wsum, aggb);
    k_dense    <<<gridD, blk, 0, stream>>>(x1, aggb, W1w, W1b, out);
}